// JumpReLUSAE_30820685316795
// MI455X (gfx1250) — compile-verified
//
#include <hip/hip_runtime.h>

// JumpReLU SAE forward on MI455X (gfx1250, wave32).
//   encode: pre = (x - b_dec) @ W_enc + b_enc ; feat = pre * (pre > thr)
//   decode: recon = feat @ W_dec + b_dec
// Memory-bound (~900MB min traffic => ~39us @ 23.3TB/s). fp32 WMMA
// (V_WMMA_F32_16X16X4_F32) keeps full reference precision with compute far
// under the bandwidth roofline. Staging uses GLOBAL_LOAD_ASYNC_TO_LDS_B128
// (ASYNCcnt-tracked, VGPR-bypassing) with LDS double buffering so the HBM
// pipe stays saturated while WMMAs run.

typedef __attribute__((ext_vector_type(2))) float v2f;
typedef __attribute__((ext_vector_type(8))) float v8f;
typedef int v4i __attribute__((vector_size(16)));  // b128 payload type for async-LDS builtin

#define BM 128
#define BN 128
#define BK 32
#define LDS_A_STRIDE 36   // 36*m mod 64 distinct for m=0..15 -> conflict-free A frags
#define LDS_B_STRIDE 136  // 2*136 mod 64 = 16 -> K/K+2 half-wave rows on disjoint banks

#if defined(__AMDGCN__) && __has_builtin(__builtin_amdgcn_global_load_async_to_lds_b128)
#define USE_ASYNC_LDS 1
#else
#define USE_ASYNC_LDS 0
#endif

#if defined(__AMDGCN__) && __has_builtin(__builtin_amdgcn_s_wait_asynccnt)
#define WAIT_ASYNC(n) __builtin_amdgcn_s_wait_asynccnt((unsigned short)(n))
#elif defined(__AMDGCN__)
#define WAIT_ASYNC(n) asm volatile("s_wait_asynccnt %0" ::"n"(n) : "memory")
#else
#define WAIT_ASYNC(n) ((void)0)
#endif

#define AS_G __attribute__((address_space(1)))
#define AS_L __attribute__((address_space(3)))

template <bool SUB_BDEC, bool JUMPRELU>
__global__ __launch_bounds__(256) void sae_gemm_wmma(
    const float* __restrict__ A,        // [Mrows x K], row stride = K
    const float* __restrict__ B,        // [K x Nout],  row stride = Nout
    const float* __restrict__ bias,     // [Nout] added in epilogue (b_enc or b_dec)
    const float* __restrict__ thresh,   // [Nout] JumpReLU threshold (encode only)
    const float* __restrict__ bdec_sub, // [K] subtracted from A columns (encode only)
    float* __restrict__ Out,            // [Mrows x Nout]
    int K, int Nout) {
  __shared__ float As[2][BM][LDS_A_STRIDE];
  __shared__ float Bs[2][BK][LDS_B_STRIDE];

  const int tid  = threadIdx.x;
  const int wave = tid >> 5;
  const int lane = tid & 31;
  const int half = lane >> 4;   // which 16-lane half
  const int l16  = lane & 15;
  const int wm   = wave & 3;    // 4 M-slots of 32 rows
  const int wn   = wave >> 2;   // 2 N-slots of 64 cols

  const int row0 = blockIdx.y * BM;
  const int col0 = blockIdx.x * BN;

  // Issue one KB tile's global->LDS copies (8 x b128 per thread, ASYNCcnt +8).
  auto stage = [&](int kt, int buf) {
    const int k0 = kt * BK;
#pragma unroll
    for (int i = 0; i < 4; ++i) {                // A tile [BM x BK]
      int f  = tid + i * 256;                    // 1024 float4 slots
      int r  = f >> 3;                           // 8 float4 per row
      int c4 = (f & 7) << 2;
      const float* gp = &A[(size_t)(row0 + r) * K + (k0 + c4)];
      float*       lp = &As[buf][r][c4];
#if USE_ASYNC_LDS
      __builtin_amdgcn_global_load_async_to_lds_b128(
          (AS_G v4i*)gp, (AS_L v4i*)lp, 0, 0);
#else
      *reinterpret_cast<float4*>(lp) = *reinterpret_cast<const float4*>(gp);
#endif
    }
#pragma unroll
    for (int i = 0; i < 4; ++i) {                // B tile [BK x BN]
      int f  = tid + i * 256;
      int r  = f >> 5;                           // 32 float4 per row
      int c4 = (f & 31) << 2;
      const float* gp = &B[(size_t)(k0 + r) * Nout + (col0 + c4)];
      float*       lp = &Bs[buf][r][c4];
#if USE_ASYNC_LDS
      __builtin_amdgcn_global_load_async_to_lds_b128(
          (AS_G v4i*)gp, (AS_L v4i*)lp, 0, 0);
#else
      *reinterpret_cast<float4*>(lp) = *reinterpret_cast<const float4*>(gp);
#endif
    }
  };

  v8f acc[2][4];
#pragma unroll
  for (int mi = 0; mi < 2; ++mi)
#pragma unroll
    for (int ni = 0; ni < 4; ++ni) {
      v8f z = {0.f, 0.f, 0.f, 0.f, 0.f, 0.f, 0.f, 0.f};
      acc[mi][ni] = z;
    }

  const int nk = K / BK;
  stage(0, 0);  // prologue: tile 0 -> buffer 0

  for (int t = 0; t < nk; ++t) {
    const int cur = t & 1;
    if (t + 1 < nk) {           // uniform branch: EXEC stays all-ones for WMMA
      stage(t + 1, cur ^ 1);
      WAIT_ASYNC(8);            // in-order: <=8 outstanding => tile t landed
    } else {
      WAIT_ASYNC(0);
    }
    __syncthreads();            // tile t visible to all waves

    const float(*At)[LDS_A_STRIDE] = As[cur];
    const float(*Bt)[LDS_B_STRIDE] = Bs[cur];
    const int k0 = t * BK;

    // b_dec slice for fused (x - b_dec): tiny (8KB total), L0/L2-resident.
    v2f bdk[8];
    if (SUB_BDEC) {
#pragma unroll
      for (int j = 0; j < 8; ++j) {
        int kidx = k0 + 4 * j + 2 * half;
        bdk[j].x = bdec_sub[kidx];
        bdk[j].y = bdec_sub[kidx + 1];
      }
    }

    // 8 k-microsteps x (2x4) tiles of V_WMMA_F32_16X16X4_F32
#pragma unroll
    for (int kk = 0; kk < BK; kk += 4) {
      v2f af[2];
      v2f bf[4];
#pragma unroll
      for (int mi = 0; mi < 2; ++mi) {
        // A 16x4 f32 layout: lanes hold M=l16; VGPR0 = K 2*half, VGPR1 = K 2*half+1
        int mrow = wm * 32 + mi * 16 + l16;
        af[mi].x = At[mrow][kk + 2 * half + 0];
        af[mi].y = At[mrow][kk + 2 * half + 1];
        if (SUB_BDEC) af[mi] -= bdk[kk >> 2];
      }
#pragma unroll
      for (int ni = 0; ni < 4; ++ni) {
        // B 4x16 f32 layout: lanes hold N=l16; VGPR0 = K 2*half, VGPR1 = K 2*half+1
        int ncol = wn * 64 + ni * 16 + l16;
        bf[ni].x = Bt[kk + 2 * half + 0][ncol];
        bf[ni].y = Bt[kk + 2 * half + 1][ncol];
      }
#pragma unroll
      for (int mi = 0; mi < 2; ++mi)
#pragma unroll
        for (int ni = 0; ni < 4; ++ni)
          acc[mi][ni] = __builtin_amdgcn_wmma_f32_16x16x4_f32(
              /*neg_a=*/false, af[mi], /*neg_b=*/false, bf[ni],
              /*c_mod=*/(short)0, acc[mi][ni],
              /*reuse_a=*/false, /*reuse_b=*/false);
    }
    __syncthreads();  // all reads of buffer `cur` done before next overwrite
  }

  // ---- epilogue: bias (+ JumpReLU) fused, direct global store ----
#pragma unroll
  for (int mi = 0; mi < 2; ++mi) {
#pragma unroll
    for (int ni = 0; ni < 4; ++ni) {
      int gcol  = col0 + wn * 64 + ni * 16 + l16;
      float bv  = bias[gcol];
      float thr = JUMPRELU ? thresh[gcol] : 0.f;
#pragma unroll
      for (int r = 0; r < 8; ++r) {
        // C/D 16x16 f32 layout: VGPR r -> M = r + 8*half, N = l16
        int grow = row0 + wm * 32 + mi * 16 + r + 8 * half;
        float v  = acc[mi][ni][r] + bv;
        if (JUMPRELU) v = (v > thr) ? v : 0.0f;
        Out[(size_t)grow * Nout + gcol] = v;
      }
    }
  }
}

extern "C" void kernel_launch(void* const* d_in, const int* in_sizes, int n_in,
                              void* d_out, int out_size, void* d_ws, size_t ws_size,
                              hipStream_t stream) {
  (void)in_sizes; (void)n_in; (void)out_size; (void)d_ws; (void)ws_size;

  const int N_ROWS = 8192, D_MODEL = 2048, D_SAE = 16384;

  const float* x     = (const float*)d_in[0];  // [8192, 2048]
  const float* W_enc = (const float*)d_in[1];  // [2048, 16384]
  const float* b_enc = (const float*)d_in[2];  // [16384]
  const float* thr   = (const float*)d_in[3];  // [16384]
  const float* W_dec = (const float*)d_in[4];  // [16384, 2048]
  const float* b_dec = (const float*)d_in[5];  // [2048]

  float* out   = (float*)d_out;
  float* recon = out;                            // [8192, 2048] first
  float* feats = out + (size_t)N_ROWS * D_MODEL; // [8192, 16384] second

  dim3 blk(256);

  // encode + JumpReLU -> features
  dim3 genc(D_SAE / BN, N_ROWS / BM);   // 128 x 64
  sae_gemm_wmma<true, true><<<genc, blk, 0, stream>>>(
      x, W_enc, b_enc, thr, b_dec, feats, D_MODEL, D_SAE);

  // decode -> reconstruction (same stream => ordered after encode)
  dim3 gdec(D_MODEL / BN, N_ROWS / BM); // 16 x 64
  sae_gemm_wmma<false, false><<<gdec, blk, 0, stream>>>(
      feats, W_dec, b_dec, nullptr, nullptr, recon, D_SAE, D_MODEL);
}